// GSplat_88364657148403
// MI455X (gfx1250) — compile-verified
//
#include <hip/hip_runtime.h>

// GSplat fused rasterizer for gfx1250 (MI455X).
//  - gs_setup: per-Gaussian params + exact grid-min of quadratic form (qmin)
//  - gs_main : params staged to LDS via TENSOR_LOAD_TO_LDS (TDM) + s_wait_tensorcnt,
//              z via V_WMMA_F32_16X16X4_F32 (features [y^2,y,1,0]) per bilinear tap,
//              v_exp_f32, integer edge masks, color accumulate, lane reduce.

typedef float v2f  __attribute__((ext_vector_type(2)));
typedef float v8f  __attribute__((ext_vector_type(8)));
typedef unsigned int u32x4v __attribute__((ext_vector_type(4)));
typedef int   i32x8v __attribute__((ext_vector_type(8)));
typedef int   i32x4v __attribute__((ext_vector_type(4)));

#define NG     1024
#define HH     256
#define WW     256
#define DELTA  (10.0f / 255.0f)
#define INVDEL 25.5f

// Params layout per Gaussian (16 floats):
// 0 ia, 1 ib, 2 id, 3 fsx, 4 fsy, 5 lnwx0, 6 lnwx1, 7 lnwy0, 8 lnwy1,
// 9 hq(=0.5*qmin), 10 c0, 11 c1, 12 c2, 13-15 pad

__global__ void gs_setup(const float* __restrict__ scale_raw,
                         const float* __restrict__ rot_raw,
                         const float* __restrict__ mean_raw,
                         const float* __restrict__ color_raw,
                         const float* __restrict__ alpha_raw,
                         float* __restrict__ P)
{
    __shared__ float sabc[3];
    __shared__ float sred[256];
    const int n = blockIdx.x;
    const int t = threadIdx.x;

    if (t == 0) {
        float s0 = 1.f / (1.f + __expf(-scale_raw[2*n + 0]));
        float s1 = 1.f / (1.f + __expf(-scale_raw[2*n + 1]));
        s0 *= s0; s1 *= s1;
        float rot = tanhf(rot_raw[n]) * 1.5707963267948966f;
        float cr = cosf(rot), sr = sinf(rot);
        float a = cr*cr*s0 + sr*sr*s1;
        float b = cr*sr*(s0 - s1);
        float d = sr*sr*s0 + cr*cr*s1;
        float det = a*d - b*b;
        float ia  =  d / det;
        float ib  = -b / det;
        float idd =  a / det;

        float sx = tanhf(mean_raw[2*n + 0]) * 127.5f;
        float sy = tanhf(mean_raw[2*n + 1]) * 127.5f;
        float fsx = floorf(sx), fsy = floorf(sy);
        float wx = sx - fsx,    wy = sy - fsy;

        float alpha = 1.f / (1.f + __expf(-alpha_raw[n]));

        float* p = P + n*16;
        p[0] = ia;  p[1] = ib;  p[2] = idd;
        p[3] = fsx; p[4] = fsy;
        p[5] = __logf(1.f - wx); p[6] = __logf(wx);
        p[7] = __logf(1.f - wy); p[8] = __logf(wy);
        p[10] = (1.f / (1.f + __expf(-color_raw[3*n + 0]))) * alpha;
        p[11] = (1.f / (1.f + __expf(-color_raw[3*n + 1]))) * alpha;
        p[12] = (1.f / (1.f + __expf(-color_raw[3*n + 2]))) * alpha;
        p[13] = 0.f; p[14] = 0.f; p[15] = 0.f;
        sabc[0] = ia; sabc[1] = ib; sabc[2] = idd;
    }
    __syncthreads();

    // Exact min over the 256x256 grid of q(x,y) = ia x^2 + 2 ib x y + id y^2.
    // For fixed x (row t), q is convex in y: grid-min is at the two grid
    // points bracketing y* = -ib*x/id (clamped).
    const float ia = sabc[0], ib = sabc[1], idd = sabc[2];
    const float x  = -5.f + (float)t * DELTA;
    const float ys = -(ib * x) / idd;
    const float jf = floorf((ys + 5.f) * INVDEL);
    const float j0 = fminf(fmaxf(jf,       0.f), 255.f);
    const float j1 = fminf(fmaxf(jf + 1.f, 0.f), 255.f);
    const float y0 = -5.f + j0 * DELTA;
    const float y1 = -5.f + j1 * DELTA;
    const float qx = ia * x * x;
    const float q0 = qx + 2.f*ib*x*y0 + idd*y0*y0;
    const float q1 = qx + 2.f*ib*x*y1 + idd*y1*y1;
    sred[t] = fminf(q0, q1);
    __syncthreads();
    for (int s = 128; s > 0; s >>= 1) {
        if (t < s) sred[t] = fminf(sred[t], sred[t + s]);
        __syncthreads();
    }
    if (t == 0) P[n*16 + 9] = 0.5f * sred[0];
}

__global__ void gs_main(const float* __restrict__ P, float* __restrict__ out)
{
    extern __shared__ float4 sp4[];   // 1024 Gaussians * 64B = 64 KB (LDS offset 0)

#if defined(__gfx1250__) && __has_builtin(__builtin_amdgcn_tensor_load_to_lds)
    // Tensor Data Mover: one 64KB 1D tile Global->LDS, issued by wave 0 only
    // (TDM ignores EXEC; wave-level branch prevents duplicate DMAs).
    if (threadIdx.x < 32) {
        const unsigned long long ga = (unsigned long long)P;
        // D# group 0: count=1 | lds_addr | global_addr[56:0] | type=2
        u32x4v g0;
        g0.x = 1u;                                    // count=1, user mode
        g0.y = 0u;                                    // lds_addr (dynamic LDS base)
        g0.z = (unsigned)ga;                          // global_addr[31:0]
        g0.w = (unsigned)((ga >> 32) & 0x1FFFFFFu) | (2u << 30);  // [56:32], type=2
        // D# group 1: data_size=4B; tensor_dim0=tile_dim0=16384 elems; dim1=1.
        i32x8v g1;
        g1[0] = 0x20000;                 // wg_mask=0 | data_size=2 (4B) << 16
        g1[1] = (int)(16384u << 16);     // tensor_dim0[15:0]  at bits [63:48]
        g1[2] = (int)(1u << 16);         // dim0[31:16]=0 | tensor_dim1[15:0]=1
        g1[3] = (int)(16384u << 16);     // dim1[31:16]=0 | tile_dim0=16384
        g1[4] = 1;                       // tile_dim1=1 | tile_dim2=0
        g1[5] = 16384;                   // tensor_dim0_stride[31:0]
        g1[6] = 0;                       // stride0[47:32]=0 | stride1[15:0]=0
        g1[7] = 0;                       // stride1[47:16]=0
        i32x4v gz4 = {0, 0, 0, 0};       // groups 2/3: dims 2-4 unused
        i32x8v gz8 = {0, 0, 0, 0, 0, 0, 0, 0};  // extra group (clang-23 6-arg form)
        __builtin_amdgcn_tensor_load_to_lds(g0, g1, gz4, gz4, gz8, 0);
        __builtin_amdgcn_s_wait_tensorcnt(0);
    }
#else
    const float4* g4 = (const float4*)P;
    for (int i = threadIdx.x; i < NG * 4; i += blockDim.x)
        sp4[i] = g4[i];
#endif
    __syncthreads();

    const int lane = threadIdx.x & 31;
    const int wave = threadIdx.x >> 5;
    const int h    = blockIdx.x >> 1;              // 512 blocks, 2 per row
    const int w0   = (blockIdx.x & 1) * 128 + wave * 16;

    const int   nl    = lane & 15;                 // Gaussian column / pixel M index
    const int   half  = lane >> 4;                 // 0: K0/K1 + M0-7, 1: K2/K3 + M8-15
    const float hf    = (float)h;
    const float w0f   = (float)w0;
    const int   wbase = w0 + half * 8;             // first pixel column of my D rows

    // A-matrix (16 pixels x K=4, f32): features [y^2, y, 1, 0].
    // ISA layout: lanes 0-15 -> (K0,K1), lanes 16-31 -> (K2,K3), M = lane&15.
    const float ym = -5.f + (w0f + (float)nl) * DELTA;
    v2f A;
    A.x = half ? 1.f : ym * ym;
    A.y = half ? 0.f : ym;

    float acc[24];
    #pragma unroll
    for (int i = 0; i < 24; i++) acc[i] = 0.f;

    #pragma unroll 1
    for (int g = 0; g < NG; g += 16) {
        const float4* pp = &sp4[(g + nl) * 4];
        const float4 p0 = pp[0];   // ia ib id fsx
        const float4 p1 = pp[1];   // fsy lnwx0 lnwx1 lnwy0
        const float4 p2 = pp[2];   // lnwy1 hq c0 c1
        const float4 p3 = pp[3];   // c2 - - -

        const float ia = p0.x, ib = p0.y, idd = p0.z, fsx = p0.w;
        const float fsy = p1.x;
        const float lnwx[2] = { p1.y, p1.z };
        const float lnwy[2] = { p1.w, p2.x };
        const float hq = p2.y;
        const float c0 = p2.z, c1 = p2.w, c2 = p3.x;
        const int   ifsx = (int)fsx;               // exact (fsx is an integer float)

        const float b0 = -0.5f * idd;              // y^2 coefficient (tap-invariant)

        float E[8];
        #pragma unroll
        for (int r = 0; r < 8; r++) E[r] = 0.f;

        #pragma unroll
        for (int ty = 0; ty < 2; ty++) {
            const float rowf  = hf + fsy + (float)ty;
            const bool  rowok = (rowf >= 0.f) && (rowf <= 255.f);
            const float xt    = -5.f + rowf * DELTA;
            #pragma unroll
            for (int tx = 0; tx < 2; tx++) {
                const float eps = (fsx + (float)tx) * DELTA;
                const float b1  = -(idd * eps + ib * xt);
                float b2 = -0.5f * (ia*xt*xt + 2.f*ib*xt*eps + idd*eps*eps)
                           + hq + lnwy[ty] + lnwx[tx];
                if (!rowok) b2 = -1e30f;           // off-image tap row -> exp = 0

                // B-matrix (K=4 x 16 Gaussians, f32):
                // lanes 0-15 -> (K0,K1), lanes 16-31 -> (K2,K3), N = lane&15.
                v2f B;
                B.x = half ? b2 : b0;
                B.y = half ? 0.f : b1;

                v8f Cz = {0.f,0.f,0.f,0.f,0.f,0.f,0.f,0.f};
                v8f Z = __builtin_amdgcn_wmma_f32_16x16x4_f32(
                            false, A, false, B, (short)0, Cz, false, false);

                // D layout: element (M = r + 8*half, N = lane&15).
                const int cb = wbase + ifsx + tx;  // tap column index base
                #pragma unroll
                for (int r = 0; r < 8; r++) {
                    float e = __expf(Z[r]);
                    e = ((unsigned)(cb + r) <= 255u) ? e : 0.f;
                    E[r] += e;
                }
            }
        }

        #pragma unroll
        for (int r = 0; r < 8; r++) {
            acc[r*3 + 0] += E[r] * c0;
            acc[r*3 + 1] += E[r] * c1;
            acc[r*3 + 2] += E[r] * c2;
        }
    }

    // Sum over the 16 Gaussian columns (lanes) inside each 16-lane half.
    #pragma unroll
    for (int i = 0; i < 24; i++) {
        float v = acc[i];
        v += __shfl_xor(v, 1, 32);
        v += __shfl_xor(v, 2, 32);
        v += __shfl_xor(v, 4, 32);
        v += __shfl_xor(v, 8, 32);
        acc[i] = v;
    }

    if (nl == 0) {                                  // lanes 0 and 16 write
        #pragma unroll
        for (int r = 0; r < 8; r++) {
            const int w    = w0 + half * 8 + r;
            const int base = (h * WW + w) * 3;
            out[base + 0] = fminf(fmaxf(acc[r*3 + 0], 0.f), 1.f);
            out[base + 1] = fminf(fmaxf(acc[r*3 + 1], 0.f), 1.f);
            out[base + 2] = fminf(fmaxf(acc[r*3 + 2], 0.f), 1.f);
        }
    }
}

extern "C" void kernel_launch(void* const* d_in, const int* in_sizes, int n_in,
                              void* d_out, int out_size, void* d_ws, size_t ws_size,
                              hipStream_t stream)
{
    (void)in_sizes; (void)n_in; (void)out_size; (void)ws_size;
    const float* scale_raw = (const float*)d_in[0];
    const float* rot_raw   = (const float*)d_in[1];
    const float* mean_raw  = (const float*)d_in[2];
    const float* color_raw = (const float*)d_in[3];
    const float* alpha_raw = (const float*)d_in[4];
    float* P = (float*)d_ws;                       // 1024*16 floats = 64 KB scratch

    gs_setup<<<NG, 256, 0, stream>>>(scale_raw, rot_raw, mean_raw,
                                     color_raw, alpha_raw, P);
    gs_main<<<(HH * WW) / 128, 256, NG * 16 * sizeof(float), stream>>>(
        P, (float*)d_out);
}